// GlobalInteractionModule_36447092474550
// MI455X (gfx1250) — compile-verified
//
#include <hip/hip_runtime.h>
#include <math.h>

typedef __attribute__((ext_vector_type(2))) float v2f;
typedef __attribute__((ext_vector_type(8))) float v8f;

#define TAU    5.0f
#define NB     32
#define NF     4096
#define ND     768
#define NCH    8        // chunks (workgroups) per batch
#define NWAVE  8        // waves per workgroup
#define NTILE  4        // 16-frame tiles per wave  (8*8*4*16 = 4096 frames)
#define WS_STRIDE 772   // floats per partial record: [M, L, acc[768], pad, pad]

// ---------------------------------------------------------------------------
// Pass 1: per-wave online softmax-weighted accumulation over 64 frames,
// sims computed with V_WMMA_F32_16X16X4_F32 (A = 16 frames x 4 dims of video,
// B = text value broadcast across all 16 columns, C accumulates over 192
// K-chunks -> C column holds the 16 frame sims).  Waves combine in LDS and
// emit one (M, L, acc[768]) record per workgroup into d_ws.
// ---------------------------------------------------------------------------
__global__ __launch_bounds__(256)
void gim_pass1(const float* __restrict__ video,
               const float* __restrict__ text,
               float* __restrict__ ws)
{
    __shared__ __align__(16) float ldsText[ND];
    __shared__ float sm[NWAVE], sl[NWAVE];
    __shared__ __align__(16) float sacc[NWAVE][ND];

    const int b    = blockIdx.x / NCH;
    const int ch   = blockIdx.x % NCH;
    const int tid  = threadIdx.x;
    const int w    = tid >> 5;
    const int lane = tid & 31;

    // Stage text[b] (3 KB) into LDS once per workgroup.
    if (tid < ND / 4) {
        *(float4*)&ldsText[4 * tid] =
            *(const float4*)(text + (size_t)b * ND + 4 * tid);
    }
    __syncthreads();

    // WMMA f32 16x16x4 operand mapping (wave32):
    //   A: lanes 0-15 -> M=lane, K={0,1} in V0/V1; lanes 16-31 -> M=lane-16, K={2,3}
    //   B: lanes 0-15 -> K={0,1} in V0/V1;         lanes 16-31 -> K={2,3}
    const int kb   = (lane >> 4) << 1;  // 0 or 2
    const int mrow = lane & 15;         // frame within tile

    float  m = -INFINITY;
    float  l = 0.0f;
    float4 acc[6];
#pragma unroll
    for (int j = 0; j < 6; ++j) acc[j] = make_float4(0.f, 0.f, 0.f, 0.f);

    for (int t = 0; t < NTILE; ++t) {
        const int frame0 = ch * (NF / NCH) + w * (NF / NCH / NWAVE) + t * 16;

        // ---- sims via WMMA: two independent accumulators to break the chain
        const float* pA = video + ((size_t)b * NF + frame0 + mrow) * ND + kb;
        const float* pT = ldsText + kb;

        v8f c0 = {};
        v8f c1 = {};
#pragma unroll 8
        for (int i = 0; i < ND / 4; i += 2) {
            v2f a0 = *(const v2f*)(pA + 4 * i);
            v2f b0 = *(const v2f*)(pT + 4 * i);
            c0 = __builtin_amdgcn_wmma_f32_16x16x4_f32(
                    false, a0, false, b0, (short)0, c0, false, false);
            v2f a1 = *(const v2f*)(pA + 4 * (i + 1));
            v2f b1 = *(const v2f*)(pT + 4 * (i + 1));
            c1 = __builtin_amdgcn_wmma_f32_16x16x4_f32(
                    false, a1, false, b1, (short)0, c1, false, false);
        }
        v8f cs = c0 + c1;

        // Every column of C is identical (B broadcast): pull the 16 sims.
        float s[16];
#pragma unroll
        for (int j = 0; j < 8; ++j) {
            s[j]     = __shfl(cs[j], 0, 32)  * (1.0f / TAU);  // M = j
            s[j + 8] = __shfl(cs[j], 16, 32) * (1.0f / TAU);  // M = j + 8
        }

        // ---- online softmax update (uniform across the wave)
        float tmax = s[0];
#pragma unroll
        for (int j = 1; j < 16; ++j) tmax = fmaxf(tmax, s[j]);
        const float m_new = fmaxf(m, tmax);
        const float scale = expf(m - m_new);   // 0 on first tile (m = -inf)

        float wv[16];
        float wsum = 0.f;
#pragma unroll
        for (int j = 0; j < 16; ++j) {
            wv[j] = expf(s[j] - m_new);
            wsum += wv[j];
        }
        l = l * scale + wsum;
        m = m_new;
#pragma unroll
        for (int j = 0; j < 6; ++j) {
            acc[j].x *= scale; acc[j].y *= scale;
            acc[j].z *= scale; acc[j].w *= scale;
        }

        // ---- weighted accumulation; tile (48 KB) re-read hits L2
        const float* pV = video + ((size_t)b * NF + frame0) * ND;
#pragma unroll
        for (int f = 0; f < 16; ++f) {
            const float wf = wv[f];
#pragma unroll
            for (int j = 0; j < 6; ++j) {
                float4 v4 = *(const float4*)(pV + (size_t)f * ND + 4 * (lane + 32 * j));
                acc[j].x = fmaf(wf, v4.x, acc[j].x);
                acc[j].y = fmaf(wf, v4.y, acc[j].y);
                acc[j].z = fmaf(wf, v4.z, acc[j].z);
                acc[j].w = fmaf(wf, v4.w, acc[j].w);
            }
        }
    }

    // ---- combine the 8 waves of this workgroup in LDS
    if (lane == 0) { sm[w] = m; sl[w] = l; }
#pragma unroll
    for (int j = 0; j < 6; ++j)
        *(float4*)&sacc[w][4 * (lane + 32 * j)] = acc[j];
    __syncthreads();

    float M = sm[0];
#pragma unroll
    for (int i = 1; i < NWAVE; ++i) M = fmaxf(M, sm[i]);
    float ew[NWAVE];
    float L = 0.f;
#pragma unroll
    for (int i = 0; i < NWAVE; ++i) {
        ew[i] = expf(sm[i] - M);
        L    += sl[i] * ew[i];
    }

    float* rec = ws + (size_t)blockIdx.x * WS_STRIDE;
    if (tid == 0) { rec[0] = M; rec[1] = L; }
#pragma unroll
    for (int r = 0; r < 3; ++r) {
        const int d = tid + 256 * r;
        float v = 0.f;
#pragma unroll
        for (int i = 0; i < NWAVE; ++i) v = fmaf(ew[i], sacc[i][d], v);
        rec[2 + d] = v;
    }
}

// ---------------------------------------------------------------------------
// Pass 2: merge the 8 chunk partials of each batch -> out[b, :]
// ---------------------------------------------------------------------------
__global__ __launch_bounds__(256)
void gim_pass2(const float* __restrict__ ws, float* __restrict__ out)
{
    const int b   = blockIdx.x;
    const int tid = threadIdx.x;

    float Mc[NCH], Lc[NCH];
#pragma unroll
    for (int c = 0; c < NCH; ++c) {
        const float* rec = ws + (size_t)(b * NCH + c) * WS_STRIDE;
        Mc[c] = rec[0];
        Lc[c] = rec[1];
    }
    float M = Mc[0];
#pragma unroll
    for (int c = 1; c < NCH; ++c) M = fmaxf(M, Mc[c]);
    float ew[NCH];
    float L = 0.f;
#pragma unroll
    for (int c = 0; c < NCH; ++c) {
        ew[c] = expf(Mc[c] - M);
        L    += Lc[c] * ew[c];
    }
    const float invL = 1.0f / L;

#pragma unroll
    for (int r = 0; r < 3; ++r) {
        const int d = tid + 256 * r;
        float v = 0.f;
#pragma unroll
        for (int c = 0; c < NCH; ++c)
            v = fmaf(ew[c], ws[(size_t)(b * NCH + c) * WS_STRIDE + 2 + d], v);
        out[(size_t)b * ND + d] = v * invL;
    }
}

extern "C" void kernel_launch(void* const* d_in, const int* in_sizes, int n_in,
                              void* d_out, int out_size, void* d_ws, size_t ws_size,
                              hipStream_t stream)
{
    const float* video = (const float*)d_in[0];  // [32, 4096, 768] f32
    const float* text  = (const float*)d_in[1];  // [32, 768] f32
    float* out = (float*)d_out;                  // [32, 768] f32
    float* ws  = (float*)d_ws;                   // needs 256*772*4 ≈ 0.77 MB

    gim_pass1<<<dim3(NB * NCH), dim3(256), 0, stream>>>(video, text, ws);
    gim_pass2<<<dim3(NB), dim3(256), 0, stream>>>(ws, out);
}